// lgn_tau_frame_86363202388406
// MI455X (gfx1250) — compile-verified
//
#include <hip/hip_runtime.h>
#include <hip/hip_bf16.h>

// ---------------------------------------------------------------------------
// LightGCN-style 3-hop propagation for MI455X (gfx1250, wave32).
// Memory/atomic bound: embedding table (76.8MB) is L2-resident (192MB L2),
// so gather + f32 atomics run out of L2. LayerNorm reductions use the CDNA5
// matrix pipe: V_WMMA_F32_16X16X4_F32 with A=ones gives exact f32 row sums
// for 16 rows per wave with zero cross-lane traffic.
// ---------------------------------------------------------------------------

#define N_USERS   100000
#define N_ITEMS   200000
#define N_TOTAL   300000
#define DIM       64
#define NNZ_E     4000000
#define N_HOPS    3
#define EPS_PERT  0.03f
#define LN_EPS    1e-5f

// d_out layout (floats): [emb_users 100000*4*64][emb_items 200000*4*64][cl_users 100000*64][cl_items 200000*64]
#define EMB_ITEM_BASE  ((size_t)N_USERS * 4 * DIM)                 // 25,600,000
#define CL_USER_BASE   ((size_t)N_TOTAL * 4 * DIM)                 // 76,800,000
#define CL_ITEM_BASE   (CL_USER_BASE + (size_t)N_USERS * DIM)      // 83,200,000

typedef float v2f __attribute__((ext_vector_type(2)));
typedef float v4f __attribute__((ext_vector_type(4)));
typedef float v8f __attribute__((ext_vector_type(8)));

// ---- stateless counter-based RNG (host+device: host derives per-hop seeds) --
__host__ __device__ __forceinline__ uint32_t hash32(uint32_t x) {
  x ^= x >> 16; x *= 0x7feb352du;
  x ^= x >> 15; x *= 0x846ca68bu;
  x ^= x >> 16;
  return x;
}
__host__ __device__ __forceinline__ float u01(uint32_t h) {
  return (float)(h >> 8) * (1.0f / 16777216.0f);   // [0,1)
}

// ---------------------------------------------------------------------------
// init: agg = concat(user, item); also write emb hop-0 slot into d_out.
// One thread per float4. Fully coalesced b128 traffic.
// ---------------------------------------------------------------------------
__global__ __launch_bounds__(256) void init_kernel(
    const float* __restrict__ uemb, const float* __restrict__ iemb,
    float* __restrict__ agg, float* __restrict__ out)
{
  size_t i = (size_t)blockIdx.x * blockDim.x + threadIdx.x;   // float4 index
  if (i >= (size_t)N_TOTAL * (DIM / 4)) return;
  size_t node = i >> 4;            // 16 float4 per row
  int    q    = (int)(i & 15);
  size_t fidx = i * 4;             // float index into concat table
  v4f v;
  if (node < N_USERS) v = *(const v4f*)(uemb + fidx);
  else                v = *(const v4f*)(iemb + (fidx - (size_t)N_USERS * DIM));
  *(v4f*)(agg + fidx) = v;
  float* embrow = (node < N_USERS)
      ? out + node * 4 * DIM
      : out + EMB_ITEM_BASE + (node - N_USERS) * 4 * DIM;
  *(v4f*)(embrow + q * 4) = v;     // hop slot 0
}

// ---------------------------------------------------------------------------
// zero the scatter destination
// ---------------------------------------------------------------------------
__global__ __launch_bounds__(256) void zero_kernel(float* __restrict__ dst)
{
  size_t i = (size_t)blockIdx.x * blockDim.x + threadIdx.x;
  if (i >= (size_t)N_TOTAL * (DIM / 4)) return;
  v4f z = {};
  *(v4f*)(dst + i * 4) = z;
}

// ---------------------------------------------------------------------------
// SpMM: one wave32 per edge, float2 per lane (64 floats/row).
// Edge dropout (p=0.5, scale 2x) decided by a per-edge hash; dropped edges
// retire immediately. Gather row from L2-resident prev, scatter with native
// global_atomic_add_f32 into L2-resident dst.
// ---------------------------------------------------------------------------
__global__ __launch_bounds__(256) void spmm_kernel(
    const float* __restrict__ prev,
    const int*   __restrict__ rows,
    const int*   __restrict__ cols,
    const float* __restrict__ vals,
    float*       __restrict__ dst,
    uint32_t seed)
{
  size_t tid = (size_t)blockIdx.x * blockDim.x + threadIdx.x;
  int e    = (int)(tid >> 5);
  int lane = (int)(threadIdx.x & 31);
  if (e >= NNZ_E) return;
  // edge_mask = floor(0.5 + U) -> keep iff U >= 0.5 ; kept edges scaled by 2
  if (u01(hash32(seed ^ (uint32_t)e)) < 0.5f) return;
  int r = rows[e];
  int c = cols[e];
  float w = vals[e] * 2.0f;
  const float* srow = prev + (size_t)c * DIM + lane * 2;
  float*       drow = dst  + (size_t)r * DIM + lane * 2;
  __builtin_prefetch(drow, 1, 3);                 // global_prefetch_b8 (write intent)
  v2f v = *(const v2f*)srow;
  __hip_atomic_fetch_add(drow + 0, w * v.x, __ATOMIC_RELAXED, __HIP_MEMORY_SCOPE_AGENT);
  __hip_atomic_fetch_add(drow + 1, w * v.y, __ATOMIC_RELAXED, __HIP_MEMORY_SCOPE_AGENT);
}

// ---------------------------------------------------------------------------
// LayerNorm + message dropout + normalized sign-noise; 16 rows per wave.
// Row sums / sums-of-squares via accumulating V_WMMA_F32_16X16X4_F32:
//   A = ones(16x4), B = data(4x16)  =>  C[m][n] = rowsum(node n), replicated,
// so every lane reads its own node's statistics from c[0] directly.
// Lane l owns node (l&15), columns 4t + 2*(l>>4) .. +1  (one float2 / chunk).
// Writes: next agg buffer, d_out emb slot hop+1, and cl slice when hop==0.
// N_TOTAL = 300000 = 16 * 18750, so every active wave is full (EXEC all-1s,
// required for WMMA).
// ---------------------------------------------------------------------------
__global__ __launch_bounds__(256) void ln_noise_kernel(
    const float* __restrict__ in,        // raw segment-sum
    float*       __restrict__ agg_out,   // normalized result (next hop input)
    float*       __restrict__ out,       // d_out base
    const float* __restrict__ gamma,
    const float* __restrict__ beta,
    int hop, uint32_t seed_drop, uint32_t seed_noise)
{
  int wid  = (int)(((size_t)blockIdx.x * blockDim.x + threadIdx.x) >> 5);
  int lane = (int)(threadIdx.x & 31);
  int node_base = wid * 16;
  if (node_base >= N_TOTAL) return;          // wave-uniform exit, EXEC stays full
  int m    = lane & 15;
  int half = lane >> 4;
  int node = node_base + m;

  const float* rowp = in + (size_t)node * DIM + 2 * half;
  v2f x[16];
#pragma unroll
  for (int t = 0; t < 16; ++t)
    x[t] = *(const v2f*)(rowp + 4 * t);

  v2f ones; ones.x = 1.0f; ones.y = 1.0f;    // A = ones(16x4)
  v8f csum = {};
  v8f csq  = {};
#pragma unroll
  for (int t = 0; t < 16; ++t) {
    csum = __builtin_amdgcn_wmma_f32_16x16x4_f32(
        false, ones, false, x[t], (short)0, csum, false, false);
    v2f xs; xs.x = x[t].x * x[t].x; xs.y = x[t].y * x[t].y;
    csq = __builtin_amdgcn_wmma_f32_16x16x4_f32(
        false, ones, false, xs, (short)0, csq, false, false);
  }
  float mean = csum[0] * (1.0f / 64.0f);
  float var  = csq[0]  * (1.0f / 64.0f) - mean * mean;
  float rstd = rsqrtf(var + LN_EPS);

  // per-node Gaussian noise (Box-Muller), pair p = column/2 keyed on node
  float nsq = 0.0f;
  v2f nz[16];
  uint32_t nk = seed_noise ^ ((uint32_t)node * 0x9E3779B9u);
#pragma unroll
  for (int t = 0; t < 16; ++t) {
    int p = (4 * t + 2 * half) >> 1;                      // pair index 0..31
    uint32_t h1 = hash32(nk + (uint32_t)p * 0x85EBCA6Bu);
    uint32_t h2 = hash32(h1 ^ 0xC2B2AE35u);
    float u1 = fmaxf(u01(h1), 1e-7f);
    float u2 = u01(h2);
    float r  = sqrtf(-2.0f * __logf(u1));
    float sn, cn;
    __sincosf(6.28318530718f * u2, &sn, &cn);
    nz[t].x = r * cn; nz[t].y = r * sn;
    nsq += nz[t].x * nz[t].x + nz[t].y * nz[t].y;
  }
  nsq += __shfl_xor(nsq, 16, 32);                         // join the two half-rows
  float nscale = EPS_PERT / (sqrtf(nsq) + 1e-12f);

  float* aggrow = agg_out + (size_t)node * DIM + 2 * half;
  float* embrow = ((node < N_USERS)
      ? out + (size_t)node * 4 * DIM
      : out + EMB_ITEM_BASE + (size_t)(node - N_USERS) * 4 * DIM)
      + (size_t)(hop + 1) * DIM + 2 * half;
  float* clrow = nullptr;
  if (hop == 0)
    clrow = ((node < N_USERS)
        ? out + CL_USER_BASE + (size_t)node * DIM
        : out + CL_ITEM_BASE + (size_t)(node - N_USERS) * DIM) + 2 * half;

  uint32_t dk = seed_drop ^ ((uint32_t)node * 0x27D4EB2Fu);
#pragma unroll
  for (int t = 0; t < 16; ++t) {
    int c = 4 * t + 2 * half;
    v2f g  = *(const v2f*)(gamma + c);
    v2f bb = *(const v2f*)(beta + c);
    float y0 = (x[t].x - mean) * rstd * g.x + bb.x;
    float y1 = (x[t].y - mean) * rstd * g.y + bb.y;
    // message dropout p=0.1, scale 1/0.9
    uint32_t h0 = hash32(dk + (uint32_t)(c + 0) * 0x9E3779B1u);
    uint32_t h1 = hash32(dk + (uint32_t)(c + 1) * 0x9E3779B1u);
    y0 = (u01(h0) >= 0.1f) ? y0 * (1.0f / 0.9f) : 0.0f;
    y1 = (u01(h1) >= 0.1f) ? y1 * (1.0f / 0.9f) : 0.0f;
    // sign(y)*unit_noise*eps  (sign(0)=0)
    float s0 = (y0 > 0.0f) ? 1.0f : ((y0 < 0.0f) ? -1.0f : 0.0f);
    float s1 = (y1 > 0.0f) ? 1.0f : ((y1 < 0.0f) ? -1.0f : 0.0f);
    y0 += s0 * nz[t].x * nscale;
    y1 += s1 * nz[t].y * nscale;
    v2f y; y.x = y0; y.y = y1;
    *(v2f*)(aggrow + 4 * t) = y;
    *(v2f*)(embrow + 4 * t) = y;
    if (clrow) *(v2f*)(clrow + 4 * t) = y;
  }
}

// ---------------------------------------------------------------------------
extern "C" void kernel_launch(void* const* d_in, const int* in_sizes, int n_in,
                              void* d_out, int out_size, void* d_ws, size_t ws_size,
                              hipStream_t stream) {
  const float* uemb  = (const float*)d_in[0];
  const float* iemb  = (const float*)d_in[1];
  const float* vals  = (const float*)d_in[2];
  const float* gamma = (const float*)d_in[3];
  const float* beta  = (const float*)d_in[4];
  const int*   adj   = (const int*)d_in[5];      // (2, NNZ): rows then cols
  const int* rows = adj;
  const int* cols = adj + NNZ_E;
  float* out = (float*)d_out;

  float* wsA = (float*)d_ws;                     // current agg (76.8 MB)
  float* wsB = wsA + (size_t)N_TOTAL * DIM;      // raw segment-sum (76.8 MB)

  const int elem4   = N_TOTAL * (DIM / 4);                 // 4.8M float4
  const int blkCopy = (elem4 + 255) / 256;
  const int blkSpmm = (int)(((size_t)NNZ_E * 32 + 255) / 256);   // 500000
  const int blkLN   = ((N_TOTAL / 16) * 32 + 255) / 256;         // 2344

  init_kernel<<<blkCopy, 256, 0, stream>>>(uemb, iemb, wsA, out);

  for (int h = 0; h < N_HOPS; ++h) {
    uint32_t se = hash32(0x45444745u + (uint32_t)h * 0x9E3779B9u);  // edge drop
    uint32_t sm = hash32(0x4D455353u + (uint32_t)h * 0x85EBCA6Bu);  // msg drop
    uint32_t sp = hash32(0x4E4F4953u + (uint32_t)h * 0xC2B2AE35u);  // noise
    zero_kernel<<<blkCopy, 256, 0, stream>>>(wsB);
    spmm_kernel<<<blkSpmm, 256, 0, stream>>>(wsA, rows, cols, vals, wsB, se);
    ln_noise_kernel<<<blkLN, 256, 0, stream>>>(wsB, wsA, out, gamma, beta, h, sm, sp);
  }
}